// MassSpringSystem_50603304682183
// MI455X (gfx1250) — compile-verified
//
#include <hip/hip_runtime.h>
#include <stdint.h>

// ---------------- problem constants (match reference) ----------------
#define N_PART  262144
#define N_SPR   1048576
#define STEPS_K 64
#define DT_K    0.01f
#define GRAV_K  9.81f
#define EPS_K   1e-6f

// ---------------- tiling ----------------
#define TPB   256          // 8 wave32s per block
#define SPT   4            // springs per thread
#define TILE  (TPB * SPT)  // 1024 springs per block -> 16KB LDS staged by TDM

// ---------------- TDM (Tensor Data Mover) helpers ----------------
typedef uint32_t v4u_t __attribute__((ext_vector_type(4)));
typedef int      v8i_t __attribute__((ext_vector_type(8)));
typedef int      v4i_t __attribute__((ext_vector_type(4)));

// Generic pointers to LDS are {SHARED_BASE, offset[31:0]} -> low 32 bits are
// the LDS byte offset the hardware (and the TDM D#) uses.
__device__ __forceinline__ uint32_t lds_offset_of(const void* p) {
    return (uint32_t)(uintptr_t)p;
}

// 1-D DMA of `ndw` dwords from global memory to LDS via TENSOR_LOAD_TO_LDS.
// D# built per CDNA5 ISA:
//   group0: count=1 (valid), lds_addr, global_addr[56:0], type=2 ("image")
//   group1: data_size=2 (4B), tensor_dim0=ndw, tensor_dim1=1,
//           tile_dim0=ndw (<=65535), tile_dim1/2 unused, dim0 stride=ndw
__device__ __forceinline__ void tdm_load_1d(uint32_t lds_off, const void* g,
                                            uint32_t ndw) {
    uint64_t ga = (uint64_t)(uintptr_t)g;
    v4u_t g0 = { 1u,                                   // count=1, is_restore=0
                 lds_off,                              // lds_addr
                 (uint32_t)ga,                         // global_addr[31:0]
                 (uint32_t)((ga >> 32) & 0x1FFFFFFull) // global_addr[56:32]
                   | (2u << 30) };                     // type=2
    v8i_t g1 = { (int)(2u << 16),                      // data_size=4B, no mask
                 (int)((ndw & 0xFFFFu) << 16),         // tensor_dim0[15:0]
                 (int)(((ndw >> 16) & 0xFFFFu)         // tensor_dim0[31:16]
                       | (1u << 16)),                  // tensor_dim1 = 1
                 (int)((ndw & 0xFFFFu) << 16),         // tile_dim0 = ndw
                 0,                                    // tile_dim1/2 unused
                 (int)ndw, 0, 0 };                     // dim0_stride = ndw
    v4i_t gz = { 0, 0, 0, 0 };
#if defined(__clang_major__) && (__clang_major__ >= 23)
    v8i_t gz8 = { 0, 0, 0, 0, 0, 0, 0, 0 };
    __builtin_amdgcn_tensor_load_to_lds(g0, g1, gz, gz, gz8, 0);
#else
    __builtin_amdgcn_tensor_load_to_lds(g0, g1, gz, gz, 0);
#endif
}

__device__ __forceinline__ void atomic_add_f32(float* p, float v) {
    __hip_atomic_fetch_add(p, v, __ATOMIC_RELAXED, __HIP_MEMORY_SCOPE_AGENT);
}

// ---------------- kernels (spring kernel FIRST so it shows in the disasm
// snippet: we want to verify tensor_load_to_lds + global_atomic_add_f32) ----
__global__ void __launch_bounds__(TPB)
ms_spring_kernel(const int* __restrict__ ij,       // E*2 (i,j pairs)
                 const float* __restrict__ rest,   // E
                 const float* __restrict__ stiff,  // E
                 const float2* __restrict__ pos,
                 float* __restrict__ force)        // N*2, accumulated
{
    __shared__ int2  s_ij[TILE];       // 8 KB
    __shared__ float s_rest[TILE];     // 4 KB
    __shared__ float s_stiff[TILE];    // 4 KB

    const int base = blockIdx.x * TILE;

    // Wave 0 issues three async TDM DMAs (EXEC ignored; TENSORcnt per-wave),
    // waits for them, then the workgroup barrier publishes LDS to all waves.
    if (threadIdx.x < 32) {
        tdm_load_1d(lds_offset_of(s_ij),    ij    + (size_t)base * 2, TILE * 2);
        tdm_load_1d(lds_offset_of(s_rest),  rest  + base,             TILE);
        tdm_load_1d(lds_offset_of(s_stiff), stiff + base,             TILE);
        __builtin_amdgcn_s_wait_tensorcnt(0);
    }
    __syncthreads();

#pragma unroll
    for (int s = 0; s < SPT; ++s) {
        const int t  = threadIdx.x + s * TPB;
        const int2 pr = s_ij[t];           // ds_load_b64
        const int pi = pr.x;
        const int pj = pr.y;
        const float L0 = s_rest[t];
        const float k  = s_stiff[t];

        const float2 xi = pos[pi];
        const float2 xj = pos[pj];
        const float dx = xj.x - xi.x;
        const float dy = xj.y - xi.y;
        const float dist = sqrtf(fmaf(dx, dx, dy * dy));
        const float coef = k * (dist - L0) / (dist + EPS_K);
        const float fx = coef * dx;
        const float fy = coef * dy;

        atomic_add_f32(&force[2 * pi + 0],  fx);
        atomic_add_f32(&force[2 * pi + 1],  fy);
        atomic_add_f32(&force[2 * pj + 0], -fx);
        atomic_add_f32(&force[2 * pj + 1], -fy);
    }
}

__global__ void __launch_bounds__(TPB)
ms_integrate_kernel(float2* __restrict__ pos, float2* __restrict__ vel,
                    float2* __restrict__ force) {
    int i = blockIdx.x * blockDim.x + threadIdx.x;
    if (i >= N_PART) return;
    float2 f = force[i];
    force[i] = make_float2(0.f, 0.f);       // re-zero for next step
    float2 v = vel[i];
    float2 p = pos[i];
    v.x += f.x * DT_K;
    v.y  = v.y + f.y * DT_K - GRAV_K * DT_K;  // gravity folded (MASS == 1)
    p.x += v.x * DT_K;
    p.y += v.y * DT_K;
    if (i == 0) { p = make_float2(0.f, 0.f); v = make_float2(0.f, 0.f); }
    pos[i] = p;
    vel[i] = v;
}

__global__ void __launch_bounds__(TPB)
ms_init_kernel(const float* __restrict__ rest_pos,   // (N-1)*2
               const float* __restrict__ vel_in,     // N*2
               float2* __restrict__ pos, float2* __restrict__ vel,
               float2* __restrict__ force) {
    int i = blockIdx.x * blockDim.x + threadIdx.x;
    if (i >= N_PART) return;
    if (i == 0) {
        pos[0] = make_float2(0.f, 0.f);
    } else {
        const float2* rp = (const float2*)rest_pos;
        pos[i] = rp[i - 1];
    }
    vel[i]   = ((const float2*)vel_in)[i];
    force[i] = make_float2(0.f, 0.f);
}

// ---------------- host side ----------------
extern "C" void kernel_launch(void* const* d_in, const int* in_sizes, int n_in,
                              void* d_out, int out_size, void* d_ws,
                              size_t ws_size, hipStream_t stream) {
    const float* rest_pos = (const float*)d_in[0];  // (N-1)*2
    const float* vel0     = (const float*)d_in[1];  // N*2
    const float* rest     = (const float*)d_in[2];  // E
    const float* stiff    = (const float*)d_in[3];  // E
    const int*   ij       = (const int*)d_in[4];    // E*2
    (void)in_sizes; (void)n_in; (void)out_size; (void)ws_size;

    float2* pos   = (float2*)d_out;                 // N*2 floats, final output
    float2* vel   = (float2*)d_ws;                  // N*2 floats
    float2* force = vel + N_PART;                   // N*2 floats

    const int pblocks = N_PART / TPB;               // 1024
    const int sblocks = N_SPR / TILE;               // 1024

    ms_init_kernel<<<pblocks, TPB, 0, stream>>>(rest_pos, vel0, pos, vel, force);

    for (int s = 0; s < STEPS_K; ++s) {
        ms_spring_kernel<<<sblocks, TPB, 0, stream>>>(ij, rest, stiff, pos,
                                                      (float*)force);
        ms_integrate_kernel<<<pblocks, TPB, 0, stream>>>(pos, vel, force);
    }
}